// GCNConv_3693671875305
// MI455X (gfx1250) — compile-verified
//
#include <hip/hip_runtime.h>
#include <hip/hip_bf16.h>

#define NIN 128
#define NOUT 128

typedef float v2f __attribute__((ext_vector_type(2)));
typedef float v8f __attribute__((ext_vector_type(8)));

// ---------------------------------------------------------------------------
// Guaranteed-native f32 atomic add (CDNA5 GLOBAL_ATOMIC_ADD_F32, no-return
// form -> STOREcnt; s_endpgm's implicit wait-idle covers completion).
// Device scope: multiple workgroups RMW the same cache lines.
// ---------------------------------------------------------------------------
__device__ __forceinline__ void atomic_add_f32_dev(float* p, float v) {
    asm volatile("global_atomic_add_f32 %0, %1, off scope:SCOPE_DEV"
                 :: "v"(p), "v"(v)
                 : "memory");
}

// ---------------------------------------------------------------------------
// Degree / normalization kernels
// ---------------------------------------------------------------------------
__global__ __launch_bounds__(256) void gcn_deg_init_kernel(float* __restrict__ deg, int n) {
    int i = blockIdx.x * 256 + threadIdx.x;
    if (i < n) deg[i] = 1.0f;  // self-loop contributes 1 to every node's degree
}

__global__ __launch_bounds__(256) void gcn_deg_count_kernel(const long long* __restrict__ dst,
                                                            float* __restrict__ deg,
                                                            int n_edges) {
    int i = blockIdx.x * 256 + threadIdx.x;
    if (i < n_edges) {
        atomic_add_f32_dev(&deg[(int)dst[i]], 1.0f);  // exact: counts < 2^24
    }
}

__global__ __launch_bounds__(256) void gcn_dinv_kernel(float* __restrict__ deg_dinv, int n) {
    int i = blockIdx.x * 256 + threadIdx.x;
    if (i < n) deg_dinv[i] = rsqrtf(deg_dinv[i]);  // deg >= 1 always
}

// ---------------------------------------------------------------------------
// h = x @ W via V_WMMA_F32_16X16X4_F32 (exact f32 path), fused epilogue:
//   h[r][n]   = (x@W)[r][n]
//   out[r][n] = b[n] + dinv[r]^2 * h[r][n]     (self-loop term + bias, inits out)
// One block = 16-row strip, 8 waves; wave w owns output columns [16w, 16w+16).
// ---------------------------------------------------------------------------
__global__ __launch_bounds__(256) void gcn_gemm_kernel(const float* __restrict__ x,
                                                       const float* __restrict__ W,
                                                       const float* __restrict__ bias,
                                                       const float* __restrict__ dinv,
                                                       float* __restrict__ h,
                                                       float* __restrict__ out) {
    __shared__ float xs[16 * NIN];   // 8 KB row-major x strip
    __shared__ float ds[16];         // dinv for the 16 rows

    const int row0 = blockIdx.x * 16;
    const int tid  = threadIdx.x;

    // Cooperative strip load: 2048 floats via float4, 2 iters per thread.
    const float4* xsrc = (const float4*)(x + (size_t)row0 * NIN);
    float4* xdst = (float4*)xs;
#pragma unroll
    for (int i = tid; i < (16 * NIN) / 4; i += 256) xdst[i] = xsrc[i];
    if (tid < 16) ds[tid] = dinv[row0 + tid];
    __syncthreads();

    const int wave = tid >> 5;
    const int lane = tid & 31;
    const int m    = lane & 15;        // A-matrix row handled by this lane
    const int kh   = lane >> 4;        // K half: lanes 16-31 hold K+2 / K+3
    const int n    = (wave << 4) | m;  // output column for B/C/D frags

    const float* xrow = xs + m * NIN + 2 * kh;
    const float* wcol = W + (size_t)(2 * kh) * NOUT + n;

    v8f acc = {};
#pragma unroll
    for (int ks = 0; ks < NIN / 4; ++ks) {
        v2f a, b;
        a.x = xrow[4 * ks];                       // A[m][4ks+2kh]
        a.y = xrow[4 * ks + 1];                   // A[m][4ks+2kh+1]
        b.x = wcol[(size_t)(4 * ks) * NOUT];      // B[4ks+2kh][n]
        b.y = wcol[(size_t)(4 * ks + 1) * NOUT];  // B[4ks+2kh+1][n]
        acc = __builtin_amdgcn_wmma_f32_16x16x4_f32(false, a, false, b,
                                                    (short)0, acc, false, false);
    }

    const float bn = bias[n];
#pragma unroll
    for (int j = 0; j < 8; ++j) {
        const int   r  = j + 8 * kh;              // C/D: VGPR j -> row j (+8 hi half)
        const float di = ds[r];
        const float v  = acc[j];
        h[(size_t)(row0 + r) * NOUT + n]   = v;
        out[(size_t)(row0 + r) * NOUT + n] = bn + di * di * v;
    }
}

// ---------------------------------------------------------------------------
// Edge scatter: one wave32 per edge; lane handles 4 consecutive channels.
// out[dst] += dinv[src]*dinv[dst] * h[src]   (f32 atomics, L2-resident)
// ---------------------------------------------------------------------------
__global__ __launch_bounds__(256) void gcn_scatter_kernel(const long long* __restrict__ src,
                                                          const long long* __restrict__ dst,
                                                          const float* __restrict__ h,
                                                          const float* __restrict__ dinv,
                                                          float* __restrict__ out,
                                                          int n_edges) {
    const int gid  = blockIdx.x * 256 + threadIdx.x;
    const int e    = gid >> 5;
    const int lane = gid & 31;
    if (e >= n_edges) return;  // wave-uniform guard

    const int   s    = (int)src[e];
    const int   d    = (int)dst[e];
    const float norm = dinv[s] * dinv[d];

    const float4 v = ((const float4*)(h + (size_t)s * NOUT))[lane];  // 512B/wave gather
    float* o = out + (size_t)d * NOUT + lane * 4;
    atomic_add_f32_dev(o + 0, norm * v.x);
    atomic_add_f32_dev(o + 1, norm * v.y);
    atomic_add_f32_dev(o + 2, norm * v.z);
    atomic_add_f32_dev(o + 3, norm * v.w);
}

// ---------------------------------------------------------------------------
extern "C" void kernel_launch(void* const* d_in, const int* in_sizes, int n_in,
                              void* d_out, int out_size, void* d_ws, size_t ws_size,
                              hipStream_t stream) {
    const float*     x    = (const float*)d_in[0];
    const long long* ei   = (const long long*)d_in[1];  // [2, E] int64
    const float*     W    = (const float*)d_in[2];      // [NIN, NOUT]
    const float*     bias = (const float*)d_in[3];      // [NOUT]
    float*           out  = (float*)d_out;

    const int n_nodes = in_sizes[0] / NIN;  // 100000 (multiple of 16)
    const int n_edges = in_sizes[1] / 2;    // 1600000

    const long long* e_src = ei;
    const long long* e_dst = ei + n_edges;

    // Workspace layout: h [n_nodes*NOUT f32] | deg->dinv [n_nodes f32]
    char*  ws   = (char*)d_ws;
    float* h    = (float*)ws;
    float* dinv = (float*)(ws + (size_t)n_nodes * NOUT * sizeof(float));

    const int nblk = (n_nodes + 255) / 256;
    gcn_deg_init_kernel<<<nblk, 256, 0, stream>>>(dinv, n_nodes);

    const int eblk = (n_edges + 255) / 256;
    gcn_deg_count_kernel<<<eblk, 256, 0, stream>>>(e_dst, dinv, n_edges);

    gcn_dinv_kernel<<<nblk, 256, 0, stream>>>(dinv, n_nodes);

    gcn_gemm_kernel<<<n_nodes / 16, 256, 0, stream>>>(x, W, bias, dinv, h, out);

    const int sblk = (int)(((long long)n_edges * 32 + 255) / 256);  // 1 wave per edge
    gcn_scatter_kernel<<<sblk, 256, 0, stream>>>(e_src, e_dst, h, dinv, out, n_edges);
}